// WaveletParsingLayer_78769700208776
// MI455X (gfx1250) — compile-verified
//
#include <hip/hip_runtime.h>
#include <hip/hip_bf16.h>

// Per-row stream compaction, [128,16,16384] f32 -> [128,16,12288] f32.
// Memory-bound: ~224 MiB total traffic -> ~10us floor at 23.3 TB/s.
// CDNA5 path: TDM async global->LDS double-buffer + s_wait_tensorcnt,
// wave32 shuffle scan for order-preserving ranks.

#define FILLER   10.1f
#define ROWS     (128 * 16)
#define L        16384
#define KEEP     12288
#define CHUNK    4096                 // floats per chunk
#define NCHUNK   (L / CHUNK)          // 4
#define CHUNK_B  (CHUNK * 4)          // 16384 bytes
#define SMEM_B   (2 * CHUNK_B + 64)   // two buffers + wave-total scratch

typedef __attribute__((ext_vector_type(4))) unsigned int v4u;
typedef __attribute__((ext_vector_type(8))) int          v8i;
typedef __attribute__((ext_vector_type(4))) int          v4i;

#if __has_builtin(__builtin_amdgcn_tensor_load_to_lds)
#define HAVE_TDM 1
#else
#define HAVE_TDM 0
#endif

#if HAVE_TDM
// Build a 1-D Tensor-DMA descriptor (ISA cdna5 ch.8) and issue
// tensor_load_to_lds: CHUNK f32 elements, global -> LDS at lds_byte_off.
__device__ __forceinline__ void tdm_load_chunk(const float* gsrc,
                                               unsigned lds_byte_off) {
    unsigned long long ga = (unsigned long long)(uintptr_t)gsrc;
    v4u g0;
    g0.x = 1u;                                   // count=1, is_restore=0, gather off
    g0.y = lds_byte_off;                         // D#.lds_addr [63:32]
    g0.z = (unsigned)(ga & 0xFFFFFFFFu);         // global_addr[31:0]  -> bits 95:64
    g0.w = (unsigned)((ga >> 32) & 0x01FFFFFFu)  // global_addr[56:32]
         | 0x80000000u;                          // type=2 ("image") in bits 127:126
    v8i g1;
    g1[0] = 0x00020000;                          // data_size=2 (4 bytes), no pad/iterate
    g1[1] = (int)((unsigned)CHUNK << 16);        // tensor_dim0[15:0] in [31:16]
    g1[2] = 0x00010000;                          // tensor_dim0 hi=0, tensor_dim1=1
    g1[3] = (int)((unsigned)CHUNK << 16);        // tile_dim0 = CHUNK
    g1[4] = 1;                                   // tile_dim1=1, tile_dim2=0
    g1[5] = CHUNK;                               // tensor_dim0_stride
    g1[6] = 0;                                   // stride hi / dim1_stride lo
    g1[7] = 0;
    v4i gz4 = {0, 0, 0, 0};                      // groups 2/3 unused (<=2D tensor)
    v8i gz8 = {0, 0, 0, 0, 0, 0, 0, 0};          // extra group (6-arg toolchain form)
    __builtin_amdgcn_tensor_load_to_lds(g0, g1, gz4, gz4, gz8, 0);
}
#endif

__global__ __launch_bounds__(256)
void compact_rows_kernel(const float* __restrict__ x3, float* __restrict__ out) {
    extern __shared__ char smem_raw[];          // dynamic-only => LDS offset 0
    float*    smem    = (float*)smem_raw;       // [2][CHUNK] staging buffers
    unsigned* warpTot = (unsigned*)(smem_raw + 2 * CHUNK_B);

    const int    row  = blockIdx.x;
    const float* rsrc = x3 + (size_t)row * L;
    float*       rdst = out + (size_t)row * KEEP;

    const unsigned tid  = threadIdx.x;
    const unsigned lane = tid & 31u;
    const unsigned wid  = tid >> 5;

#if HAVE_TDM
    if (tid == 0) {
        tdm_load_chunk(rsrc, 0u);               // prologue: chunk 0 -> buf 0
    }
#endif

    int running = 0;                            // row-level write offset (uniform)
    for (int c = 0; c < NCHUNK; ++c) {
#if HAVE_TDM
        __syncthreads();                        // buf[(c+1)&1] no longer being read
        if (tid == 0) {
            if (c + 1 < NCHUNK) {
                tdm_load_chunk(rsrc + (size_t)(c + 1) * CHUNK,
                               (unsigned)(((c + 1) & 1) * CHUNK_B));
                __builtin_amdgcn_s_wait_tensorcnt(1);   // chunk c landed
            } else {
                __builtin_amdgcn_s_wait_tensorcnt(0);
            }
        }
        __syncthreads();                        // chunk c visible to all waves
        const float* buf = smem + (size_t)(c & 1) * CHUNK;
#else
        const float* buf = rsrc + (size_t)c * CHUNK;   // direct coalesced reads
#endif
        // Each thread owns elements e = s*1024 + 4*tid + j, s=0..3, j=0..3.
        float4   q[4];
        unsigned packed = 0;                    // 4 per-sub-chunk counts, byte-packed
        #pragma unroll
        for (int s = 0; s < 4; ++s) {
            q[s] = *reinterpret_cast<const float4*>(buf + s * 1024 + 4 * tid);
            unsigned cnt = (unsigned)(q[s].x != FILLER) + (unsigned)(q[s].y != FILLER)
                         + (unsigned)(q[s].z != FILLER) + (unsigned)(q[s].w != FILLER);
            packed |= cnt << (8 * s);
        }

        // wave32 inclusive scan of byte-packed counts (per-byte max 128: no carry)
        unsigned incl = packed;
        #pragma unroll
        for (int d = 1; d < 32; d <<= 1) {
            unsigned n = (unsigned)__shfl_up((int)incl, (unsigned)d, 32);
            if (lane >= (unsigned)d) incl += n;
        }
        const unsigned excl = incl - packed;
        if (lane == 31u) warpTot[wid] = incl;   // wave totals, byte-packed
        __syncthreads();

        // combine 8 wave totals (unpacked: block totals can exceed 255)
        unsigned totAll[4]   = {0, 0, 0, 0};
        unsigned exclWave[4] = {0, 0, 0, 0};
        #pragma unroll
        for (int w = 0; w < 8; ++w) {
            unsigned v = warpTot[w];
            #pragma unroll
            for (int s = 0; s < 4; ++s) {
                unsigned cmp = (v >> (8 * s)) & 0xFFu;
                totAll[s] += cmp;
                if ((unsigned)w < wid) exclWave[s] += cmp;
            }
        }

        // scatter survivors in row order
        int subBase = running;
        #pragma unroll
        for (int s = 0; s < 4; ++s) {
            int pos = subBase + (int)exclWave[s] + (int)((excl >> (8 * s)) & 0xFFu);
            float vals[4] = {q[s].x, q[s].y, q[s].z, q[s].w};
            #pragma unroll
            for (int j = 0; j < 4; ++j) {
                float v = vals[j];
                if (v != FILLER) {
                    if (pos < KEEP) rdst[pos] = v;
                    ++pos;
                }
            }
            subBase += (int)totAll[s];
        }
        running = subBase;                      // identical in every thread
    }
}

extern "C" void kernel_launch(void* const* d_in, const int* in_sizes, int n_in,
                              void* d_out, int out_size, void* d_ws, size_t ws_size,
                              hipStream_t stream) {
    // setup_inputs order: x1, x2, x3, keep_len -> only x3 matters.
    const float* x3  = (const float*)d_in[2];
    float*       out = (float*)d_out;
    hipLaunchKernelGGL(compact_rows_kernel, dim3(ROWS), dim3(256), SMEM_B, stream,
                       x3, out);
}